// MultiHeadAttentionQuantum_65481071395567
// MI455X (gfx1250) — compile-verified
//
#include <hip/hip_runtime.h>
#include <math.h>

// Problem constants (match reference)
#define B_ 8
#define S_ 4096
#define E_ 8
// softmax run in base-2 domain: fold SCALE * log2(e) into the score scale
static constexpr float SCALE_LOG2E_ = 0.5f * 1.4426950408889634f;

typedef float v2f __attribute__((ext_vector_type(2)));
typedef float v8f __attribute__((ext_vector_type(8)));

#if __has_builtin(__builtin_amdgcn_exp2f)
#define EXP2F(x) __builtin_amdgcn_exp2f(x)
#else
#define EXP2F(x) exp2f(x)
#endif

// Swap the two 16-lane halves of the wave (lane i <-> lane i^16).
// v_permlanex16_b32 with identity selects is a pure-VALU alternative to
// ds_bpermute: no LDS round trip, no s_wait_dscnt in the hot loop.
__device__ __forceinline__ float xor16(float x) {
#if __has_builtin(__builtin_amdgcn_permlanex16)
    int i = __builtin_amdgcn_permlanex16(__float_as_int(x), __float_as_int(x),
                                         0x76543210, 0xfedcba98,
                                         /*fi=*/true, /*bound_ctrl=*/false);
    return __int_as_float(i);
#else
    return __shfl_xor(x, 16, 32);
#endif
}

// ---------------------------------------------------------------------------
// Kernel 1: fused q/k/v projection + quantum RX measurement
//   q = cos(x @ Wq^T + bq + theta), same for k, v.
// One thread per (b,s) row; weights staged in LDS.
// V is stored transposed (vT[b][e][s]) so the attention kernel's V^T
// A-fragments become contiguous b64 loads.
// ---------------------------------------------------------------------------
__global__ __launch_bounds__(256)
void qkv_proj_kernel(const float* __restrict__ x,
                     const float* __restrict__ Wq, const float* __restrict__ bq,
                     const float* __restrict__ Wk, const float* __restrict__ bk,
                     const float* __restrict__ Wv, const float* __restrict__ bv,
                     const float* __restrict__ theta,
                     float* __restrict__ qbuf, float* __restrict__ kbuf,
                     float* __restrict__ vT)
{
    __shared__ float sW[3 * 64];
    __shared__ float sB[4 * 8]; // bq, bk, bv, theta
    const int tid = threadIdx.x;
    if (tid < 64) {
        sW[tid]       = Wq[tid];
        sW[64 + tid]  = Wk[tid];
        sW[128 + tid] = Wv[tid];
    }
    if (tid < 8) {
        sB[tid]      = bq[tid];
        sB[8 + tid]  = bk[tid];
        sB[16 + tid] = bv[tid];
        sB[24 + tid] = theta[tid];
    }
    __syncthreads();

    const int row = blockIdx.x * blockDim.x + tid; // [0, B*S)
    if (row >= B_ * S_) return;
    const int bi = row >> 12;      // row / S_
    const int si = row & (S_ - 1); // row % S_

    const float4* xp = (const float4*)(x + (size_t)row * E_);
    float4 x0 = xp[0], x1 = xp[1];
    float xr[8] = {x0.x, x0.y, x0.z, x0.w, x1.x, x1.y, x1.z, x1.w};

#pragma unroll
    for (int m = 0; m < 3; ++m) {
        float o[8];
#pragma unroll
        for (int i = 0; i < 8; ++i) {
            float acc = sB[m * 8 + i] + sB[24 + i];
#pragma unroll
            for (int j = 0; j < 8; ++j) acc = fmaf(xr[j], sW[m * 64 + i * 8 + j], acc);
            o[i] = cosf(acc);
        }
        if (m < 2) { // q, k: row-major
            float* dst = (m == 0 ? qbuf : kbuf) + (size_t)row * E_;
            ((float4*)dst)[0] = make_float4(o[0], o[1], o[2], o[3]);
            ((float4*)dst)[1] = make_float4(o[4], o[5], o[6], o[7]);
        } else {     // v: transposed vT[b][e][s]
#pragma unroll
            for (int i = 0; i < 8; ++i)
                vT[((size_t)bi * E_ + i) * S_ + si] = o[i];
        }
    }
}

// ---------------------------------------------------------------------------
// Kernel 2: flash attention (online softmax, scores never hit memory) fused
// with the final 8x8 output projection.
//
// One wave owns 16 query rows of one batch; 32 keys per loop iteration
// (two 16-key subtiles share one max/alpha/rescale). Per iteration:
//   S^T = K_tile(16x8) @ Q^T(8x16)        -> 2x2 v_wmma_f32_16x16x4_f32
//   out^T += V^T(16x16 pad) @ P^T(16x16)  -> 2x4 v_wmma_f32_16x16x4_f32
//
// f32 WMMA fragment conventions (h = lane>>4, n = lane&15, j = vgpr 0/1):
//   A 16x4 : lane holds A[M=n][K=2h+j]
//   B 4x16 : lane holds B[K=2h+j][N=n]
//   C/D    : vgpr r holds D[M=r+8h][N=n]
// ---------------------------------------------------------------------------
__global__ __launch_bounds__(128)
void flash_attn_kernel(const float* __restrict__ qbuf,
                       const float* __restrict__ kbuf,
                       const float* __restrict__ vT,
                       const float* __restrict__ Wc,
                       const float* __restrict__ bc,
                       float* __restrict__ out)
{
    const int lane = threadIdx.x & 31;
    const int wave = threadIdx.x >> 5;
    const int h = lane >> 4; // half-wave
    const int n = lane & 15; // M/N stripe index

    const int tilesPerBatch = S_ / 64; // 64 queries per block (4 waves x 16)
    const int b = blockIdx.x / tilesPerBatch;
    const int qbase = (blockIdx.x % tilesPerBatch) * 64 + wave * 16;

    const float* qB = qbuf + (size_t)b * S_ * E_;
    const float* kB = kbuf + (size_t)b * S_ * E_;
    const int e = n & 7;
    const float amask = (n < 8) ? 1.0f : 0.0f; // V^T pad rows (E>=8) are zero
    const float* vRow = vT + ((size_t)b * E_ + e) * S_;

    // Loop-invariant B operand of GEMM1: Q^T. bqf[c][j] = q[qbase+n][4c+2h+j]
    v2f bqf[2];
#pragma unroll
    for (int c = 0; c < 2; ++c)
        bqf[c] = *(const v2f*)(qB + (size_t)(qbase + n) * E_ + 4 * c + 2 * h);

    v8f acc = {};          // out^T accumulator, D layout
    float m = -1e30f;      // running row max, base-2 scaled domain
    float l = 0.0f;        // running row sum

    for (int t0 = 0; t0 < S_; t0 += 32) {
        // prefetch the K / V^T stream a few tiles ahead (global_prefetch_b8)
        __builtin_prefetch(kB + (size_t)(t0 + 128 + n) * E_, 0, 1);
        __builtin_prefetch(vRow + t0 + 128, 0, 1);

        // ---- GEMM1 for both subtiles: S^T = K_tile @ Q^T (scaled to log2 domain)
        v8f s[2];
#pragma unroll
        for (int sub = 0; sub < 2; ++sub) {
            const int t0s = t0 + 16 * sub;
            v8f d = {};
#pragma unroll
            for (int c = 0; c < 2; ++c) {
                v2f a = *(const v2f*)(kB + (size_t)(t0s + n) * E_ + 4 * c + 2 * h);
                d = __builtin_amdgcn_wmma_f32_16x16x4_f32(false, a, false, bqf[c],
                                                          (short)0, d, false, false);
            }
#pragma unroll
            for (int r = 0; r < 8; ++r) d[r] *= SCALE_LOG2E_;
            s[sub] = d;
        }

        // ---- shared online-softmax update for the 32 keys
        float tmax = -1e30f;
#pragma unroll
        for (int sub = 0; sub < 2; ++sub)
#pragma unroll
            for (int r = 0; r < 8; ++r) tmax = fmaxf(tmax, s[sub][r]);
        tmax = fmaxf(tmax, xor16(tmax));
        const float mnew = fmaxf(m, tmax);
        const float alpha = EXP2F(m - mnew);
        float psum = 0.0f;
#pragma unroll
        for (int sub = 0; sub < 2; ++sub)
#pragma unroll
            for (int r = 0; r < 8; ++r) {
                s[sub][r] = EXP2F(s[sub][r] - mnew); // now holds P^T
                psum += s[sub][r];
            }
        psum += xor16(psum);
        l = l * alpha + psum;
        m = mnew;
#pragma unroll
        for (int r = 0; r < 8; ++r) acc[r] *= alpha;

        // ---- GEMM2 per subtile: out^T += V^T @ P^T over the 16 keys
#pragma unroll
        for (int sub = 0; sub < 2; ++sub) {
            const int t0s = t0 + 16 * sub;
            const v8f p = s[sub];
            // cross-half exchange: send exactly what the partner half needs
            // (h=0 lanes need partner p[0,1,4,5]; h=1 lanes need partner p[2,3,6,7])
            float recv[4];
            recv[0] = xor16(h ? p[0] : p[2]);
            recv[1] = xor16(h ? p[1] : p[3]);
            recv[2] = xor16(h ? p[4] : p[6]);
            recv[3] = xor16(h ? p[5] : p[7]);

#pragma unroll
            for (int c = 0; c < 4; ++c) {
                // A chunk: V^T rows (E), keys 4c+2h .. 4c+2h+1 — contiguous in vT
                v2f a = *(const v2f*)(vRow + t0s + 4 * c + 2 * h);
                a[0] *= amask;
                a[1] *= amask;
                // B chunk: P^T[4c+2h+j][n] from own regs or the exchanged set
                v2f bp;
                if (c == 0) { bp[0] = h ? recv[0] : p[0]; bp[1] = h ? recv[1] : p[1]; }
                if (c == 1) { bp[0] = h ? recv[2] : p[4]; bp[1] = h ? recv[3] : p[5]; }
                if (c == 2) { bp[0] = h ? p[2] : recv[0]; bp[1] = h ? p[3] : recv[1]; }
                if (c == 3) { bp[0] = h ? p[6] : recv[2]; bp[1] = h ? p[7] : recv[3]; }
                acc = __builtin_amdgcn_wmma_f32_16x16x4_f32(false, a, false, bp,
                                                            (short)0, acc, false, false);
            }
        }
    }

    // ---- finalize: half-wave 0 lanes hold the whole row out[q=n][E=r] in acc[r]
    if (h == 0) {
        const float invl = 1.0f / l;
        float o[8];
#pragma unroll
        for (int r = 0; r < 8; ++r) o[r] = acc[r] * invl;

        // fused output projection: res = o @ Wc^T + bc
        float res[8];
#pragma unroll
        for (int i = 0; i < 8; ++i) {
            float a2 = bc[i];
#pragma unroll
            for (int j = 0; j < 8; ++j) a2 = fmaf(o[j], Wc[i * 8 + j], a2);
            res[i] = a2;
        }
        float4* op = (float4*)(out + ((size_t)b * S_ + qbase + n) * E_);
        op[0] = make_float4(res[0], res[1], res[2], res[3]);
        op[1] = make_float4(res[4], res[5], res[6], res[7]);
    }
}

// ---------------------------------------------------------------------------
// Launch
// ---------------------------------------------------------------------------
extern "C" void kernel_launch(void* const* d_in, const int* in_sizes, int n_in,
                              void* d_out, int out_size, void* d_ws, size_t ws_size,
                              hipStream_t stream) {
    (void)in_sizes; (void)n_in; (void)out_size; (void)ws_size;
    const float* x     = (const float*)d_in[0];
    const float* Wq    = (const float*)d_in[1];
    const float* bq    = (const float*)d_in[2];
    const float* Wk    = (const float*)d_in[3];
    const float* bk    = (const float*)d_in[4];
    const float* Wv    = (const float*)d_in[5];
    const float* bv    = (const float*)d_in[6];
    const float* theta = (const float*)d_in[7];
    const float* Wc    = (const float*)d_in[8];
    const float* bc    = (const float*)d_in[9];
    float* out = (float*)d_out;

    // workspace: q, k (row-major) and v (transposed), 1 MB each (f32)
    float* qbuf = (float*)d_ws;
    float* kbuf = qbuf + (size_t)B_ * S_ * E_;
    float* vT   = kbuf + (size_t)B_ * S_ * E_;

    qkv_proj_kernel<<<(B_ * S_ + 255) / 256, 256, 0, stream>>>(
        x, Wq, bq, Wk, bk, Wv, bv, theta, qbuf, kbuf, vT);

    flash_attn_kernel<<<B_ * (S_ / 64), 128, 0, stream>>>(
        qbuf, kbuf, vT, Wc, bc, out);
}